// DEVISELayer_18743237280105
// MI455X (gfx1250) — compile-verified
//
#include <hip/hip_runtime.h>
#include <hip/hip_bf16.h>

typedef float v2f __attribute__((ext_vector_type(2)));
typedef float v8f __attribute__((ext_vector_type(8)));

#define BB 4096
#define DD 1024
#define CC 20000
#define DC 64
#define MARGIN 0.1f

#define ROWB 4                              // row tiles per wave in hinge kernel
#define CT_TILES 1250                       // 20000 / 16
#define CT_CHUNK 10                         // class tiles per block
#define N_CHUNKS (CT_TILES / CT_CHUNK)      // 125

// ---------------- Kernel A: proj = X @ W  (4096x1024 @ 1024x64) ----------------
// One wave per 16x16 output tile. grid = (256, 4), block = 32.
__global__ __launch_bounds__(32)
void devise_proj_kernel(const float* __restrict__ X, const float* __restrict__ W,
                        float* __restrict__ proj) {
    const int tm   = blockIdx.x;        // 0..255 row tile
    const int tn   = blockIdx.y;        // 0..3   col tile
    const int lane = threadIdx.x;       // 0..31
    const int mn   = lane & 15;         // m for A, n for B/D
    const int half = lane >> 4;         // 0 or 1
    const int kb   = half * 2;          // k base within 4-wide step

    const float* xrow = X + (size_t)(tm * 16 + mn) * DD;

    v8f acc = {0.f, 0.f, 0.f, 0.f, 0.f, 0.f, 0.f, 0.f};
    for (int k0 = 0; k0 < DD; k0 += 4) {
        v2f a, b;
        a.x = xrow[k0 + kb];
        a.y = xrow[k0 + kb + 1];
        b.x = W[(size_t)(k0 + kb)     * DC + tn * 16 + mn];
        b.y = W[(size_t)(k0 + kb + 1) * DC + tn * 16 + mn];
        acc = __builtin_amdgcn_wmma_f32_16x16x4_f32(false, a, false, b,
                                                    (short)0, acc, false, false);
    }
    // D layout: VGPR r -> row (r + 8*half), col mn
    #pragma unroll
    for (int r = 0; r < 8; ++r) {
        proj[(size_t)(tm * 16 + r + half * 8) * DC + tn * 16 + mn] = acc[r];
    }
}

// ---------------- Kernel B: true_sim[b] = proj[b] . E[y[b]] ----------------
// grid = 4096, block = 64.
__global__ __launch_bounds__(64)
void devise_truesim_kernel(const float* __restrict__ proj, const int* __restrict__ y,
                           const float* __restrict__ E, float* __restrict__ ts) {
    const int b = blockIdx.x;
    const int t = threadIdx.x;
    const int cls = y[b];
    __shared__ float s[64];
    s[t] = proj[(size_t)b * DC + t] * E[(size_t)cls * DC + t];
    __syncthreads();
    for (int off = 32; off > 0; off >>= 1) {
        if (t < off) s[t] += s[t + off];
        __syncthreads();
    }
    if (t == 0) ts[b] = s[0];
}

// ---------------- init: zero per_sample ----------------
__global__ __launch_bounds__(256)
void devise_zero_kernel(float* __restrict__ p, int n) {
    int i = blockIdx.x * 256 + threadIdx.x;
    if (i < n) p[i] = 0.f;
}

// ---------------- Kernel C: fused sims GEMM + hinge reduction ----------------
// Each wave owns ROWB=4 row tiles (A fragments + true_sim kept in registers)
// and streams class tiles; every B fragment (E^T) is reused for 4 WMMAs.
// grid = (64 row blocks, 125 class chunks), block = 32 (one wave).
__global__ __launch_bounds__(32)
void devise_hinge_kernel(const float* __restrict__ proj, const float* __restrict__ E,
                         const float* __restrict__ ts, float* __restrict__ per_sample) {
    const int rb    = blockIdx.x;       // row block 0..63 (4 row tiles each)
    const int chunk = blockIdx.y;       // class chunk 0..124
    const int lane  = threadIdx.x;
    const int mn    = lane & 15;
    const int half  = lane >> 4;
    const int kb    = half * 2;

    // Preload A fragments for 4 row tiles x 16 k-steps (loop-invariant).
    v2f afrag[ROWB][16];
    #pragma unroll
    for (int rt = 0; rt < ROWB; ++rt) {
        const float* arow = proj + (size_t)((rb * ROWB + rt) * 16 + mn) * DC;
        #pragma unroll
        for (int ks = 0; ks < 16; ++ks) {
            afrag[rt][ks].x = arow[ks * 4 + kb];
            afrag[rt][ks].y = arow[ks * 4 + kb + 1];
        }
    }
    // Preload true_sim for the 8 rows per tile this thread's accumulators cover.
    float tsl[ROWB][8];
    #pragma unroll
    for (int rt = 0; rt < ROWB; ++rt)
        #pragma unroll
        for (int r = 0; r < 8; ++r)
            tsl[rt][r] = ts[(rb * ROWB + rt) * 16 + r + half * 8];

    float acc[ROWB][8];
    #pragma unroll
    for (int rt = 0; rt < ROWB; ++rt)
        #pragma unroll
        for (int r = 0; r < 8; ++r) acc[rt][r] = 0.f;

    const int ct0 = chunk * CT_CHUNK;
    for (int ct = ct0; ct < ct0 + CT_CHUNK; ++ct) {
        // B[k][n] = E[class n][k]; lane holds class (ct*16 + mn)
        const float* erow = E + (size_t)(ct * 16 + mn) * DC;
        v8f c[ROWB];
        #pragma unroll
        for (int rt = 0; rt < ROWB; ++rt)
            c[rt] = (v8f){0.f, 0.f, 0.f, 0.f, 0.f, 0.f, 0.f, 0.f};
        #pragma unroll
        for (int ks = 0; ks < 16; ++ks) {
            v2f b;
            b.x = erow[ks * 4 + kb];
            b.y = erow[ks * 4 + kb + 1];
            #pragma unroll
            for (int rt = 0; rt < ROWB; ++rt)
                c[rt] = __builtin_amdgcn_wmma_f32_16x16x4_f32(
                            false, afrag[rt][ks], false, b, (short)0, c[rt],
                            false, false);
        }
        #pragma unroll
        for (int rt = 0; rt < ROWB; ++rt)
            #pragma unroll
            for (int r = 0; r < 8; ++r) {
                float h = MARGIN + c[rt][r] - tsl[rt][r];
                acc[rt][r] += (h > 0.f) ? h : 0.f;
            }
    }

    // Reduce each row-sum across the 16 lanes of this half-wave, then atomic.
    #pragma unroll
    for (int rt = 0; rt < ROWB; ++rt)
        #pragma unroll
        for (int r = 0; r < 8; ++r) {
            float v = acc[rt][r];
            v += __shfl_xor(v, 1, 32);
            v += __shfl_xor(v, 2, 32);
            v += __shfl_xor(v, 4, 32);
            v += __shfl_xor(v, 8, 32);
            if (mn == 0)
                atomicAdd(&per_sample[(rb * ROWB + rt) * 16 + r + half * 8], v);
        }
}

// ---------------- Kernel D: final reduction -> mean - MARGIN ----------------
__global__ __launch_bounds__(256)
void devise_final_kernel(const float* __restrict__ per_sample, float* __restrict__ out) {
    __shared__ float s[256];
    float v = 0.f;
    for (int i = threadIdx.x; i < BB; i += 256) v += per_sample[i];
    s[threadIdx.x] = v;
    __syncthreads();
    for (int off = 128; off > 0; off >>= 1) {
        if (threadIdx.x < off) s[threadIdx.x] += s[threadIdx.x + off];
        __syncthreads();
    }
    // per_sample includes the label class (contributes exactly MARGIN each)
    if (threadIdx.x == 0) out[0] = s[0] / (float)BB - MARGIN;
}

extern "C" void kernel_launch(void* const* d_in, const int* in_sizes, int n_in,
                              void* d_out, int out_size, void* d_ws, size_t ws_size,
                              hipStream_t stream) {
    const float* X = (const float*)d_in[0];          // (4096, 1024)
    const int*   y = (const int*)d_in[1];            // (4096,)
    const float* E = (const float*)d_in[2];          // (20000, 64)
    const float* W = (const float*)d_in[3];          // (1024, 64)
    float* out = (float*)d_out;

    // Workspace layout
    float* proj       = (float*)d_ws;                               // 4096*64 = 1 MB
    float* true_sim   = proj + (size_t)BB * DC;                     // 16 KB
    float* per_sample = true_sim + BB;                              // 16 KB

    // 1) proj = X @ W
    devise_proj_kernel<<<dim3(BB / 16, DC / 16), 32, 0, stream>>>(X, W, proj);
    // 2) true_sim
    devise_truesim_kernel<<<BB, 64, 0, stream>>>(proj, y, E, true_sim);
    // 3) zero per_sample
    devise_zero_kernel<<<(BB + 255) / 256, 256, 0, stream>>>(per_sample, BB);
    // 4) fused sims + hinge (register-blocked 4 row tiles per wave)
    devise_hinge_kernel<<<dim3(BB / (16 * ROWB), N_CHUNKS), 32, 0, stream>>>(
        proj, E, true_sim, per_sample);
    // 5) final mean - margin
    devise_final_kernel<<<1, 256, 0, stream>>>(per_sample, out);
}